// LoopFallbackEval_12678743458597
// MI455X (gfx1250) — compile-verified
//
#include <hip/hip_runtime.h>

// out = x + 4096.0f, x: (16384, 4096) fp32 -> 67,108,864 elements.
// Pure HBM streaming: 536.9 MB traffic, ~23 us floor at 23.3 TB/s.
// AI = 0.125 flop/byte -> WMMA inapplicable; optimized CDNA5 path is
// 128-bit non-temporal global VMEM on wave32 with max per-wave MLP.
//
// Block-contiguous tiling: each block owns 1024 consecutive float4s (16 KiB);
// a thread's 4 accesses are 4 KiB apart -> encodable as 24-bit immediate
// offsets off one index VGPR (scale_offset), eliminating 64-bit address math.

typedef __attribute__((ext_vector_type(4))) float v4f;

#define ADD_TPB       256   // 8 wave32 waves per block
#define V4_PER_THREAD 4     // 4 x 16B = 64B per thread, 4 loads in flight
#define TILE_V4       (ADD_TPB * V4_PER_THREAD)   // 1024 float4s per block

// Branch-free kernel for exact tiling (reference shape).
__global__ __launch_bounds__(ADD_TPB) void
add_const_exact_kernel(const v4f* __restrict__ x, v4f* __restrict__ out)
{
    const int base = (int)(blockIdx.x * TILE_V4 + threadIdx.x);
    const float K = 4096.0f;

    v4f a = __builtin_nontemporal_load(&x[base]);                 // +0
    v4f b = __builtin_nontemporal_load(&x[base +     ADD_TPB]);   // +4 KiB
    v4f c = __builtin_nontemporal_load(&x[base + 2 * ADD_TPB]);   // +8 KiB
    v4f d = __builtin_nontemporal_load(&x[base + 3 * ADD_TPB]);   // +12 KiB
    a += K; b += K; c += K; d += K;
    __builtin_nontemporal_store(a, &out[base]);
    __builtin_nontemporal_store(b, &out[base +     ADD_TPB]);
    __builtin_nontemporal_store(c, &out[base + 2 * ADD_TPB]);
    __builtin_nontemporal_store(d, &out[base + 3 * ADD_TPB]);
}

// Guarded fallback for shapes that don't tile (not used for the reference).
__global__ __launch_bounds__(ADD_TPB) void
add_const_guarded_kernel(const v4f* __restrict__ x, v4f* __restrict__ out, int n4)
{
    const int base = (int)(blockIdx.x * TILE_V4 + threadIdx.x);
    const float K = 4096.0f;

#pragma unroll
    for (int k = 0; k < V4_PER_THREAD; ++k) {
        const int idx = base + k * ADD_TPB;
        if (idx < n4) {
            v4f v = __builtin_nontemporal_load(&x[idx]);
            v += K;
            __builtin_nontemporal_store(v, &out[idx]);
        }
    }
}

extern "C" void kernel_launch(void* const* d_in, const int* in_sizes, int n_in,
                              void* d_out, int out_size, void* d_ws, size_t ws_size,
                              hipStream_t stream)
{
    (void)n_in; (void)d_ws; (void)ws_size; (void)out_size;

    const v4f* x  = (const v4f*)d_in[0];
    v4f*      out = (v4f*)d_out;

    const int n  = in_sizes[0];   // 67,108,864
    const int n4 = n >> 2;        // 16,777,216 float4s

    if (n4 % TILE_V4 == 0) {
        // Reference shape: 16384 blocks, exact tiling, branch-free kernel.
        add_const_exact_kernel<<<n4 / TILE_V4, ADD_TPB, 0, stream>>>(x, out);
    } else {
        const int blocks = (n4 + TILE_V4 - 1) / TILE_V4;
        add_const_guarded_kernel<<<blocks, ADD_TPB, 0, stream>>>(x, out, n4);
    }
}